// SSMLayer_73297911873685
// MI455X (gfx1250) — compile-verified
//
#include <hip/hip_runtime.h>
#include <hip/hip_bf16.h>

typedef float v2f __attribute__((ext_vector_type(2)));
typedef float v8f __attribute__((ext_vector_type(8)));

#define LSEQ  4096
#define NHEAD 1024
#define NS    64
#define NCH   64   // LSEQ / chunk(64)

// C(64x64) = [C +] A(64x64) @ B(64x64), all row-major in LDS.
// 256 threads = 8 wave32; each wave computes two 16x16 tiles via
// V_WMMA_F32_16X16X4_F32, 16 K-steps of 4.
__device__ __forceinline__ void mm64(const float* __restrict__ A,
                                     const float* __restrict__ B,
                                     float* __restrict__ C,
                                     bool accumulate, int tid)
{
    const int wave = tid >> 5;
    const int lane = tid & 31;
    const int lh   = lane & 15;          // row/col within tile
    const int kh   = (lane >> 4) << 1;   // K sub-pair: 0 or 2
    const int rh   = (lane >> 4) << 3;   // +8 row offset for upper lanes (C/D layout)
#pragma unroll
    for (int rep = 0; rep < 2; ++rep) {
        const int tile = wave * 2 + rep;     // 0..15 over 4x4 tile grid
        const int tm = (tile >> 2) << 4;
        const int tn = (tile & 3) << 4;
        v8f acc = {};
#pragma unroll
        for (int kk = 0; kk < 16; ++kk) {
            const int kb = kk * 4 + kh;
            v2f a, b;
            // A 16x4 fragment: lanes 0-15 -> K={0,1}, lanes 16-31 -> K={2,3}
            a.x = A[(tm + lh) * NS + kb];
            a.y = A[(tm + lh) * NS + kb + 1];
            // B 4x16 fragment: row striped across lanes within VGPR
            b.x = B[kb * NS + tn + lh];
            b.y = B[(kb + 1) * NS + tn + lh];
            acc = __builtin_amdgcn_wmma_f32_16x16x4_f32(
                false, a, false, b, (short)0, acc, false, false);
        }
#pragma unroll
        for (int r = 0; r < 8; ++r) {
            const int idx = (tm + rh + r) * NS + tn + lh;
            C[idx] = (accumulate ? C[idx] : 0.0f) + acc[r];
        }
    }
}

extern "C" __global__ __launch_bounds__(256)
void ssm_head_kernel(const float* __restrict__ inp,      // (L,H)
                     const float* __restrict__ As,       // (H,64,64)
                     const float* __restrict__ Bs,       // (H,64,1)
                     const float* __restrict__ Cs,       // (H,1,64)
                     const float* __restrict__ Ds,       // (H,)
                     const float* __restrict__ logsteps, // (H,)
                     float* __restrict__ out)            // (L,H)
{
    extern __shared__ float smem[];
    const int head = blockIdx.x;
    const int tid  = threadIdx.x;

    // LDS layout (floats)
    float* G    = smem;               // 64x129 Gauss scratch (8256)
    float* bufA = smem;               // reused: power ping / Toeplitz T
    float* bufB = smem + 4096;        // reused: power pong -> holds Ab^64
    float* AbS  = smem + 8256;        // 64x64 discretized A
    float* Pr   = AbS  + 4096;        // Pr[:,i] = Ab^{63-i} x0
    float* Qm   = Pr   + 4096;        // Q[j] = C Ab^{j+1}
    float* Uc   = Qm   + 4096;        // Uc[i][c] = u[c*64+i]
    float* Vb   = Uc   + 4096;        // V; reused as OutC
    float* Sp   = Vb   + 4096;        // Sprev[:,c] = state entering chunk c
    float* x0v  = Sp   + 4096;        // 64
    float* C2v  = x0v  + 64;
    float* Kv   = C2v  + 64;
    float* pcur = Kv   + 64;
    float* qcur = pcur + 64;
    float* scur = qcur + 64;
    float* fac  = scur + 64;

    // ---- Phase 0: build augmented [ I-hA | I+hA | step*B ]  (64 x 129) ----
    const float stepv = expf(logsteps[head]);
    const float hh    = 0.5f * stepv;
    const float* Ah   = As + (size_t)head * 4096;
    for (int e = tid; e < 64 * 129; e += 256) {
        const int r = e / 129, c = e % 129;
        float v;
        if (c < 64)       v = ((c == r) ? 1.0f : 0.0f) - hh * Ah[r * 64 + c];
        else if (c < 128) v = (((c - 64) == r) ? 1.0f : 0.0f) + hh * Ah[r * 64 + (c - 64)];
        else              v = stepv * Bs[head * 64 + r];
        G[e] = v;
    }
    __syncthreads();

    // ---- Phase 1: Gauss-Jordan: solve (I-hA) [Ab | x0] = [I+hA | step*B] ----
    for (int piv = 0; piv < 64; ++piv) {
        const float inv = 1.0f / G[piv * 129 + piv];
        __syncthreads();
        if (tid < 129) G[piv * 129 + tid] *= inv;
        __syncthreads();
        if (tid < 64) fac[tid] = (tid == piv) ? 0.0f : G[tid * 129 + piv];
        __syncthreads();
        for (int e = tid; e < 64 * 129; e += 256) {
            const int r = e / 129, c = e % 129;
            if (r != piv) G[e] -= fac[r] * G[piv * 129 + c];
        }
        __syncthreads();
    }

    // ---- Phase 2: extract Ab, x0, C2; load input chunk-major ----
    if (tid < 64) {
        x0v[tid] = G[tid * 129 + 128];
        C2v[tid] = Cs[head * 64 + tid];
    }
    for (int e = tid; e < 4096; e += 256) {
        const int r = e >> 6, c = e & 63;
        AbS[e] = G[r * 129 + 64 + c];
    }
    for (int t = tid; t < LSEQ; t += 256) {
        const int c = t >> 6, i = t & 63;
        Uc[i * NS + c] = inp[(size_t)t * NHEAD + head];
    }
    __syncthreads();

    // ---- Phase 3: serial chains p_j = Ab p_{j-1} (thr 0-63),
    //               q_j = q_{j-1} Ab (thr 64-127), 64 steps each ----
    if (tid < 64) {
        pcur[tid] = x0v[tid];
        Pr[tid * NS + 63] = x0v[tid];
        qcur[tid] = C2v[tid];
    }
    __syncthreads();
    for (int j = 1; j <= 64; ++j) {
        float pn = 0.0f, qn = 0.0f;
        if (tid < 64) {
            for (int k = 0; k < NS; ++k) pn += AbS[tid * NS + k] * pcur[k];
        } else if (tid < 128) {
            const int c = tid - 64;
            for (int k = 0; k < NS; ++k) qn += qcur[k] * AbS[k * NS + c];
        }
        __syncthreads();
        if (tid < 64) {
            if (j < 64) { pcur[tid] = pn; Pr[tid * NS + (63 - j)] = pn; }
        } else if (tid < 128) {
            const int c = tid - 64;
            Qm[(j - 1) * NS + c] = qn;   // Q[j-1] = C Ab^j
            qcur[c] = qn;
        }
        __syncthreads();
    }

    // SSM kernel values K[k] = C . Ab^k x0 = C . Pr[:,63-k]
    if (tid < 64) {
        float s = 0.0f;
        for (int k = 0; k < NS; ++k) s += C2v[k] * Pr[k * NS + (63 - tid)];
        Kv[tid] = s;
    }
    __syncthreads();

    // ---- Phase 4: Ab^64 by repeated squaring (6 WMMA matmuls) ----
    mm64(AbS,  AbS,  bufA, false, tid); __syncthreads();  // Ab^2
    mm64(bufA, bufA, bufB, false, tid); __syncthreads();  // Ab^4
    mm64(bufB, bufB, bufA, false, tid); __syncthreads();  // Ab^8
    mm64(bufA, bufA, bufB, false, tid); __syncthreads();  // Ab^16
    mm64(bufB, bufB, bufA, false, tid); __syncthreads();  // Ab^32
    mm64(bufA, bufA, bufB, false, tid); __syncthreads();  // Ab^64 -> bufB

    // ---- Phase 5: chunk inputs -> state increments: V = Pr @ Uc (WMMA) ----
    mm64(Pr, Uc, Vb, false, tid); __syncthreads();

    // ---- Phase 6: inter-chunk scan: s_c = Ab^64 s_{c-1} + V[:,c] ----
    if (tid < 64) scur[tid] = 0.0f;
    __syncthreads();
    for (int c = 0; c < NCH; ++c) {
        float ns = 0.0f;
        if (tid < 64) {
            Sp[tid * NCH + c] = scur[tid];
            ns = Vb[tid * NCH + c];
            for (int k = 0; k < NS; ++k) ns += bufB[tid * NS + k] * scur[k];
        }
        __syncthreads();
        if (tid < 64) scur[tid] = ns;
        __syncthreads();
    }

    // ---- Phase 7: lower-triangular Toeplitz T[j][i] = K[j-i] ----
    for (int e = tid; e < 4096; e += 256) {
        const int j = e >> 6, i = e & 63;
        bufA[e] = (j >= i) ? Kv[j - i] : 0.0f;
    }
    __syncthreads();

    // ---- Phase 8: OutC = Q @ Sprev + T @ Uc (2 WMMA matmuls) ----
    float* OutC = Vb;  // V is dead after the scan
    mm64(Qm,   Sp, OutC, false, tid); __syncthreads();
    mm64(bufA, Uc, OutC, true,  tid); __syncthreads();

    // ---- Phase 9: out[t,h] = OutC[j][c] + D*u ----
    const float Dh = Ds[head];
    for (int e = tid; e < 4096; e += 256) {
        const int j = e >> 6, c = e & 63;
        const int t = c * 64 + j;
        out[(size_t)t * NHEAD + head] = OutC[e] + Dh * Uc[e];
    }
}

extern "C" void kernel_launch(void* const* d_in, const int* in_sizes, int n_in,
                              void* d_out, int out_size, void* d_ws, size_t ws_size,
                              hipStream_t stream) {
    const float* inp  = (const float*)d_in[0];
    const float* As   = (const float*)d_in[1];
    const float* Bs   = (const float*)d_in[2];
    const float* Cs   = (const float*)d_in[3];
    const float* Ds   = (const float*)d_in[4];
    const float* lgs  = (const float*)d_in[5];
    float* out        = (float*)d_out;

    const size_t shmem = (size_t)(8256 + 6 * 4096 + 7 * 64) * sizeof(float); // 133120 B
    hipFuncSetAttribute((const void*)ssm_head_kernel,
                        hipFuncAttributeMaxDynamicSharedMemorySize, (int)shmem);
    ssm_head_kernel<<<NHEAD, 256, shmem, stream>>>(inp, As, Bs, Cs, Ds, lgs, out);
}